// QueryAttention_87033217286188
// MI455X (gfx1250) — compile-verified
//
#include <hip/hip_runtime.h>
#include <hip/hip_bf16.h>
#include <math.h>

// ---------------------------------------------------------------------------
// Problem constants (B=32, T=1, S=4096, INDIM=OUTDIM=512), fp32 throughout.
// Memory-bound on memory_bank (256 MB): single streamed pass, online softmax.
// ---------------------------------------------------------------------------
#define ATTN_B 32
#define ATTN_S 4096
#define ATTN_D 512          // INDIM == OUTDIM
#define ROWS 64             // memory-bank rows per LDS chunk
#define SPLIT 8             // S-splits per batch (flash-decoding style)
#define CH_PER_SPLIT (ATTN_S / SPLIT / ROWS)          // 8 chunks per split
#define LOADS_PER_THREAD (ROWS * ATTN_D / 4 / 256)    // 32 float4 asyncs/thread

typedef float v2f __attribute__((ext_vector_type(2)));
typedef float v8f __attribute__((ext_vector_type(8)));

// ---------------------------------------------------------------------------
// WMMA f32 GEMM:  C[m][n] = sum_k a(m,k) * W[n*K + k]   (W row-major [N,K])
// a(m,k) = k < K0 ? A0[m*K0 + k] : A1[m*(K-K0) + (k-K0)]   (concat support)
// One wave per 16x16 C tile, V_WMMA_F32_16X16X4_F32 over K in steps of 4.
// A layout: lanes 0-15 hold K={0,1}, lanes 16-31 hold K={2,3} (2 VGPRs).
// B layout mirrors A over K; C/D: VGPR r -> row r (+8 for lanes 16-31).
// ---------------------------------------------------------------------------
__global__ __launch_bounds__(32)
void gemm_wT_wmma(const float* __restrict__ A0, const float* __restrict__ A1,
                  int K0, int K,
                  const float* __restrict__ W, float* __restrict__ C, int N) {
  const int lane = threadIdx.x;
  const int half = lane >> 4;         // 0: lanes 0-15, 1: lanes 16-31
  const int l15  = lane & 15;
  const int n = blockIdx.x * 16 + l15;  // B/C column
  const int m = blockIdx.y * 16 + l15;  // A row
  const int K1 = K - K0;

  v8f acc = {0.f, 0.f, 0.f, 0.f, 0.f, 0.f, 0.f, 0.f};

  for (int k0 = 0; k0 < K; k0 += 4) {
    const int ka = k0 + 2 * half;     // this lane's K pair
    v2f a, bv;
    if (ka < K0) {
      a.x = A0[m * K0 + ka];
      a.y = A0[m * K0 + ka + 1];
    } else {
      a.x = A1[m * K1 + (ka - K0)];
      a.y = A1[m * K1 + (ka - K0) + 1];
    }
    bv.x = W[n * K + ka];
    bv.y = W[n * K + ka + 1];
    acc = __builtin_amdgcn_wmma_f32_16x16x4_f32(
        /*neg_a=*/false, a, /*neg_b=*/false, bv,
        /*c_mod=*/(short)0, acc, /*reuse_a=*/false, /*reuse_b=*/false);
  }

  const int cm = blockIdx.y * 16 + 8 * half;
  const int cn = blockIdx.x * 16 + l15;
#pragma unroll
  for (int r = 0; r < 8; ++r)
    C[(cm + r) * N + cn] = acc[r];
}

// ---------------------------------------------------------------------------
// Async chunk loader: 64 rows x 512 f32 (128 KB) global -> LDS.
// GVS mode: saddr = SGPR pair (uniform chunk base), vaddr = per-lane byte
// offset; the 24-bit inst offset is added to BOTH global and LDS addresses.
// Generic pointers to __shared__ carry the LDS byte offset in bits [31:0].
// ---------------------------------------------------------------------------
__device__ __forceinline__ unsigned lds_lo32(const void* p) {
  return (unsigned)(unsigned long long)p;
}

__device__ __forceinline__ void issue_chunk_async(const float* gbase,
                                                  unsigned ldsBase, int t) {
  const unsigned goff  = (unsigned)(t * 16);
  const unsigned daddr = ldsBase + goff;
#pragma unroll
  for (int q = 0; q < LOADS_PER_THREAD; ++q) {
    asm volatile("global_load_async_to_lds_b128 %0, %1, %2 offset:%3"
                 :: "v"(daddr), "v"(goff), "s"(gbase), "i"(q * 4096)
                 : "memory");
  }
}

// ---------------------------------------------------------------------------
// Fused scores + online softmax + context, one HBM pass over memory_bank.
// grid = (B, SPLIT), 256 threads (8 wave32). Double-buffered LDS (264 KB).
// Emits raw scores (for the align output) and per-split partials (M, l, acc).
// ---------------------------------------------------------------------------
__global__ __launch_bounds__(256)
void attn_stream(const float* __restrict__ mem,       // [B,S,D]
                 const float* __restrict__ ht,        // [B,D]
                 const long long* __restrict__ lens,  // [B]
                 float* __restrict__ raw_scores,      // [B,S] (d_out align area)
                 float* __restrict__ pM, float* __restrict__ pL,
                 float* __restrict__ pC)              // [B,SPLIT,(1,1,D)]
{
  __shared__ __align__(16) float buf[2][ROWS * ATTN_D];   // 2 x 128 KB
  __shared__ __align__(16) float sh_h[ATTN_D];
  __shared__ float sh_s[ROWS];

  const int b  = blockIdx.x;
  const int sp = blockIdx.y;
  const int t  = threadIdx.x;
  const int wave = t >> 5, lane = t & 31;
  const long long len = lens[b];
  const int row0 = sp * (ATTN_S / SPLIT);
  const float* memb = mem + ((size_t)b * ATTN_S + row0) * ATTN_D;

  for (int i = t; i < ATTN_D; i += 256) sh_h[i] = ht[b * ATTN_D + i];

  const unsigned lb0 = lds_lo32(&buf[0][0]);
  const unsigned lb1 = lds_lo32(&buf[1][0]);

  issue_chunk_async(memb, lb0, t);   // prefetch chunk 0

  float acc0 = 0.f, acc1 = 0.f, lrun = 0.f, Mrun = -INFINITY;

  for (int ci = 0; ci < CH_PER_SPLIT; ++ci) {
    // Prefetch next chunk into the other buffer (freed by last-iter barrier).
    if (ci + 1 < CH_PER_SPLIT) {
      issue_chunk_async(memb + (size_t)(ci + 1) * ROWS * ATTN_D,
                        (ci & 1) ? lb0 : lb1, t);
      asm volatile("s_wait_asynccnt 0x20" ::: "memory");  // chunk ci landed
    } else {
      asm volatile("s_wait_asynccnt 0x0" ::: "memory");
    }
    __syncthreads();  // everyone's piece of chunk ci is in LDS

    const float* cur = &buf[ci & 1][0];

    // Phase A: scores s_j = <h, m_j>; 8 rows per wave, float4 + shfl reduce.
#pragma unroll
    for (int rr = 0; rr < ROWS / 8; ++rr) {
      const int r = wave * (ROWS / 8) + rr;
      const float4* mrow = (const float4*)(cur + r * ATTN_D);
      const float4* hp   = (const float4*)sh_h;
      float part = 0.f;
#pragma unroll
      for (int kk = 0; kk < 4; ++kk) {
        float4 mv = mrow[lane + kk * 32];
        float4 hv = hp[lane + kk * 32];
        part += mv.x * hv.x + mv.y * hv.y + mv.z * hv.z + mv.w * hv.w;
      }
#pragma unroll
      for (int off = 16; off > 0; off >>= 1) part += __shfl_xor(part, off, 32);
      if (lane == 0) {
        const int j = row0 + ci * ROWS + r;
        sh_s[r] = (j < len) ? part : -INFINITY;
        raw_scores[(size_t)b * ATTN_S + j] = part;
      }
    }
    __syncthreads();  // scores visible to all

    // Phase B: online-softmax rescale + accumulate from the same LDS chunk.
    float mchunk = -INFINITY;
#pragma unroll 8
    for (int j = 0; j < ROWS; ++j) mchunk = fmaxf(mchunk, sh_s[j]);
    if (mchunk > -INFINITY) {           // skip fully-masked chunks (no NaN)
      const float Mn = fmaxf(Mrun, mchunk);
      const float scale = __expf(Mrun - Mn);   // exp(-inf)=0 on first chunk
      acc0 *= scale; acc1 *= scale; lrun *= scale;
#pragma unroll 8
      for (int j = 0; j < ROWS; ++j) {
        const float p = __expf(sh_s[j] - Mn);  // masked -> exp(-inf)=0
        lrun += p;
        acc0 = fmaf(p, cur[j * ATTN_D + t], acc0);
        acc1 = fmaf(p, cur[j * ATTN_D + t + 256], acc1);
      }
      Mrun = Mn;
    }
    __syncthreads();  // chunk buffer free for the prefetch 2 iters ahead
  }

  const size_t pidx = (size_t)b * SPLIT + sp;
  pC[pidx * ATTN_D + t]       = acc0;
  pC[pidx * ATTN_D + t + 256] = acc1;
  if (t == 0) { pM[pidx] = Mrun; pL[pidx] = lrun; }
}

// ---------------------------------------------------------------------------
// Flash-decoding combine of the SPLIT partials -> c[b,:] and global (M, l).
// ---------------------------------------------------------------------------
__global__ __launch_bounds__(256)
void attn_combine(const float* __restrict__ pM, const float* __restrict__ pL,
                  const float* __restrict__ pC,
                  float* __restrict__ c_out, float* __restrict__ Ml_out) {
  const int b = blockIdx.x, t = threadIdx.x;
  float M = -INFINITY;
#pragma unroll
  for (int i = 0; i < SPLIT; ++i) M = fmaxf(M, pM[b * SPLIT + i]);
  float l = 0.f, a0 = 0.f, a1 = 0.f;
#pragma unroll
  for (int i = 0; i < SPLIT; ++i) {
    const float mi = pM[b * SPLIT + i];
    if (mi > -INFINITY) {               // masked split: acc=0, l=0 -> skip
      const float w = __expf(mi - M);
      l += w * pL[b * SPLIT + i];
      const float* pc = pC + ((size_t)b * SPLIT + i) * ATTN_D;
      a0 = fmaf(w, pc[t], a0);
      a1 = fmaf(w, pc[t + 256], a1);
    }
  }
  c_out[b * ATTN_D + t]       = a0 / l;
  c_out[b * ATTN_D + t + 256] = a1 / l;
  if (t == 0) { Ml_out[b * 2] = M; Ml_out[b * 2 + 1] = l; }
}

// ---------------------------------------------------------------------------
// Normalize raw scores in-place into softmax probabilities (align output).
// ---------------------------------------------------------------------------
__global__ __launch_bounds__(256)
void norm_align(float* __restrict__ align, const float* __restrict__ Ml,
                const long long* __restrict__ lens) {
  const int b = blockIdx.x, t = threadIdx.x;
  const float M = Ml[b * 2], l = Ml[b * 2 + 1];
  const long long len = lens[b];
  for (int j = t; j < ATTN_S; j += 256) {
    const float s = align[(size_t)b * ATTN_S + j];
    align[(size_t)b * ATTN_S + j] = (j < len) ? (__expf(s - M) / l) : 0.0f;
  }
}

// ---------------------------------------------------------------------------
// Host-side launch.  Inputs: source, memory_bank, memory_lengths(i64),
// W_in[512,512], W_out[512,1024].  d_out = attn_h[32*512] ++ align[32*4096].
// ---------------------------------------------------------------------------
extern "C" void kernel_launch(void* const* d_in, const int* in_sizes, int n_in,
                              void* d_out, int out_size, void* d_ws, size_t ws_size,
                              hipStream_t stream) {
  const float*     src  = (const float*)d_in[0];
  const float*     mem  = (const float*)d_in[1];
  const long long* lens = (const long long*)d_in[2];   // jnp int64
  const float*     Win  = (const float*)d_in[3];
  const float*     Wout = (const float*)d_in[4];

  float* out_attn  = (float*)d_out;                    // [32,512]
  float* out_align = out_attn + ATTN_B * ATTN_D;       // [32,4096]

  float* ws    = (float*)d_ws;
  float* ws_h  = ws;                                   // 32*512   h_t
  float* ws_c  = ws_h + ATTN_B * ATTN_D;               // 32*512   context c
  float* ws_Ml = ws_c + ATTN_B * ATTN_D;               // 32*2     global M,l
  float* ws_pM = ws_Ml + ATTN_B * 2;                   // 32*8     split max
  float* ws_pL = ws_pM + ATTN_B * SPLIT;               // 32*8     split sum
  float* ws_pC = ws_pL + ATTN_B * SPLIT;               // 32*8*512 split acc

  // 1) h_t = source @ W_in^T        (WMMA f32 16x16x4)
  gemm_wT_wmma<<<dim3(ATTN_D / 16, ATTN_B / 16), 32, 0, stream>>>(
      src, src, ATTN_D, ATTN_D, Win, ws_h, ATTN_D);

  // 2) fused scores + online softmax + context (single HBM pass, async->LDS)
  attn_stream<<<dim3(ATTN_B, SPLIT), 256, 0, stream>>>(
      mem, ws_h, lens, out_align, ws_pM, ws_pL, ws_pC);

  // 3) combine split partials
  attn_combine<<<ATTN_B, 256, 0, stream>>>(ws_pM, ws_pL, ws_pC, ws_c, ws_Ml);

  // 4) align_vectors = masked softmax(raw scores)
  norm_align<<<ATTN_B, 256, 0, stream>>>(out_align, ws_Ml, lens);

  // 5) attn_h = [c ; source] @ W_out^T   (WMMA f32 16x16x4, K=1024 concat)
  gemm_wT_wmma<<<dim3(ATTN_D / 16, ATTN_B / 16), 32, 0, stream>>>(
      ws_c, src, ATTN_D, ATTN_D * 2, Wout, out_attn, ATTN_D);
}